// DisLoss_12197707120668
// MI455X (gfx1250) — compile-verified
//
#include <hip/hip_runtime.h>
#include <hip/hip_bf16.h>
#include <math.h>

#define N_STATES 32
#define FEAT_DIM 512
#define INV_TEMP 10.0f          // 1 / SCALING_TEMPERATURE
#define EMA_EPS  1e-12f
#define PF_DIST  16             // cache-prefetch distance (rows ahead)

typedef __attribute__((ext_vector_type(2))) float v2f;
typedef __attribute__((ext_vector_type(8))) float v8f;

__device__ __forceinline__ unsigned ballot32(bool p) {
#if __has_builtin(__builtin_amdgcn_ballot_w32)
    return __builtin_amdgcn_ballot_w32(p);
#else
    return (unsigned)__ballot(p ? 1 : 0);
#endif
}

// Load one 2KB feature row fully coalesced: lane l takes contiguous elements
// [l*16, l*16+16) as four b128 loads.
#define LOAD_ROW(buf, idx)                                                      \
    do {                                                                        \
        const float4* _src =                                                    \
            reinterpret_cast<const float4*>(feats + (size_t)(idx) * FEAT_DIM) + \
            lane * 4;                                                           \
        float4 _t0 = _src[0], _t1 = _src[1], _t2 = _src[2], _t3 = _src[3];      \
        buf[0] = _t0.x;  buf[1] = _t0.y;  buf[2]  = _t0.z;  buf[3]  = _t0.w;    \
        buf[4] = _t1.x;  buf[5] = _t1.y;  buf[6]  = _t1.z;  buf[7]  = _t1.w;    \
        buf[8] = _t2.x;  buf[9] = _t2.y;  buf[10] = _t2.z;  buf[11] = _t2.w;    \
        buf[12] = _t3.x; buf[13] = _t3.y; buf[14] = _t3.z;  buf[15] = _t3.w;    \
    } while (0)

// Cache-side prefetch of a future row (global_prefetch_b8): each lane covers
// its own 64B slice, so the wave touches every cacheline of the 2KB row.
#define PF_ROW(idx)                                                             \
    __builtin_prefetch(feats + (size_t)(idx) * FEAT_DIM + lane * 16, 0, 1)

// One EMA + renormalize step on register-resident prototype p[16]/lane,
// consuming buffer `fb`, then immediately refilling `fb` with row t+4 (its
// cachelines were prefetched up to PF_DIST-4 steps ago, so the refill hits cache) so
// the load latency overlaps this step's reduction/normalize.
#define EMA_STEP(fb, t)                                                         \
    do {                                                                        \
        float v_[16];                                                           \
        float ss_ = 0.f;                                                        \
        _Pragma("unroll") for (int k = 0; k < 16; ++k) {                        \
            float vv = 0.5f * p[k] + 0.5f * fb[k];                              \
            v_[k] = vv;                                                         \
            ss_ += vv * vv;                                                     \
        }                                                                       \
        if ((t) + 4 < count) { int _nx = mylist[(t) + 4]; LOAD_ROW(fb, _nx); }  \
        if ((t) + PF_DIST < count) {                                            \
            int _pf = mylist[(t) + PF_DIST];                                    \
            PF_ROW(_pf);                                                        \
        }                                                                       \
        ss_ += __shfl_xor(ss_, 16, 32);                                         \
        ss_ += __shfl_xor(ss_, 8, 32);                                          \
        ss_ += __shfl_xor(ss_, 4, 32);                                          \
        ss_ += __shfl_xor(ss_, 2, 32);                                          \
        ss_ += __shfl_xor(ss_, 1, 32);                                          \
        float inv_ = 1.0f / fmaxf(sqrtf(ss_), EMA_EPS);                         \
        _Pragma("unroll") for (int k = 0; k < 16; ++k) p[k] = v_[k] * inv_;     \
    } while (0)

// One wave per prototype row. Phase 0: count (labels < s) and (labels == s)
// via wave32 ballots -> this state's compact-list offset + length. Phase A:
// ordered stream compaction of matching sample indices. Phase B: sequential
// EMA walk with depth-4 register pipeline + distance-16 cache prefetch.
__global__ __launch_bounds__(32) void disloss_ema_kernel(
    const float* __restrict__ feats, const float* __restrict__ protos_init,
    const int* __restrict__ labels, int batch, int* __restrict__ list,
    float* __restrict__ protos_out) {
    const int s = blockIdx.x;
    const int lane = threadIdx.x;

    // ---- Phase 0: offset = #labels<s, count = #labels==s ----
    int offs = 0, count = 0;
    for (int base = 0; base < batch; base += 32) {
        int j = base + lane;
        int lab = (j < batch) ? labels[j] : 0x7fffffff;
        offs += __popc(ballot32(lab < s));
        count += __popc(ballot32(lab == s));
    }

    // ---- Phase A: ordered compaction of this state's sample indices ----
    int wpos = 0;
    for (int base = 0; base < batch; base += 32) {
        int j = base + lane;
        int lab = (j < batch) ? labels[j] : 0x7fffffff;
        bool hit = (lab == s);
        unsigned m = ballot32(hit);
        if (hit) {
            int rank = __popc(m & ((1u << lane) - 1u));
            list[offs + wpos + rank] = j;
        }
        wpos += __popc(m);
    }

    // ---- Phase B: sequential EMA, 4-deep load pipeline ----
    float p[16];
    {
        const float4* pin =
            reinterpret_cast<const float4*>(protos_init + (size_t)s * FEAT_DIM) +
            lane * 4;
        float4 a0 = pin[0], a1 = pin[1], a2 = pin[2], a3 = pin[3];
        p[0] = a0.x;  p[1] = a0.y;  p[2] = a0.z;   p[3] = a0.w;
        p[4] = a1.x;  p[5] = a1.y;  p[6] = a1.z;   p[7] = a1.w;
        p[8] = a2.x;  p[9] = a2.y;  p[10] = a2.z;  p[11] = a2.w;
        p[12] = a3.x; p[13] = a3.y; p[14] = a3.z;  p[15] = a3.w;
    }

    const int* mylist = list + offs;
    float b0[16], b1[16], b2[16], b3[16];
    if (count > 0) { int i0 = mylist[0]; LOAD_ROW(b0, i0); }
    if (count > 1) { int i1 = mylist[1]; LOAD_ROW(b1, i1); }
    if (count > 2) { int i2 = mylist[2]; LOAD_ROW(b2, i2); }
    if (count > 3) { int i3 = mylist[3]; LOAD_ROW(b3, i3); }
    for (int q = 4; q < PF_DIST && q < count; ++q) {
        int ip = mylist[q];
        PF_ROW(ip);
    }

    int t = 0;
    while (t < count) {
        EMA_STEP(b0, t);
        ++t;
        if (t >= count) break;
        EMA_STEP(b1, t);
        ++t;
        if (t >= count) break;
        EMA_STEP(b2, t);
        ++t;
        if (t >= count) break;
        EMA_STEP(b3, t);
        ++t;
    }

    {
        float4* pout =
            reinterpret_cast<float4*>(protos_out + (size_t)s * FEAT_DIM) + lane * 4;
        pout[0] = make_float4(p[0], p[1], p[2], p[3]);
        pout[1] = make_float4(p[4], p[5], p[6], p[7]);
        pout[2] = make_float4(p[8], p[9], p[10], p[11]);
        pout[3] = make_float4(p[12], p[13], p[14], p[15]);
    }
}

// 4 waves, each computes one 16x16 tile of protos @ protos.T with full-fp32
// V_WMMA_F32_16X16X4_F32 (K=512 in steps of 4), then wave 0 does the
// off-diagonal log-mean-exp and NaN-masked mean.
__global__ __launch_bounds__(128) void disloss_wmma_loss_kernel(
    const float* __restrict__ P, float* __restrict__ out) {
    __shared__ float logits[32][33];  // padded stride -> conflict-free

    const int tid = threadIdx.x;
    const int wave = tid >> 5;
    const int lane = tid & 31;
    const int l = lane & 15;     // row-in-half
    const int hi = lane >> 4;    // which K-pair (ISA f32 16x4 A/B layout)
    const int m0 = (wave >> 1) << 4;
    const int n0 = (wave & 1) << 4;

    const float* Arow = P + (size_t)(m0 + l) * FEAT_DIM + (hi << 1);
    const float* Brow = P + (size_t)(n0 + l) * FEAT_DIM + (hi << 1);

    v8f c = {};
#pragma unroll 4
    for (int kk = 0; kk < FEAT_DIM; kk += 4) {
        v2f a;
        a.x = Arow[kk];
        a.y = Arow[kk + 1];
        v2f b;
        b.x = Brow[kk];
        b.y = Brow[kk + 1];
        // D = A * B + C, fp32 throughout (matches reference precision)
        c = __builtin_amdgcn_wmma_f32_16x16x4_f32(
            /*neg_a=*/false, a, /*neg_b=*/false, b,
            /*c_mod=*/(short)0, c, /*reuse_a=*/false, /*reuse_b=*/false);
    }

    // C layout (ISA 7.12.2): VGPR r holds row m0 + 8*hi + r, col n0 + l.
#pragma unroll
    for (int r = 0; r < 8; ++r) {
        int row = m0 + (hi << 3) + r;
        int col = n0 + l;
        logits[row][col] = c[r] * INV_TEMP;
    }
    __syncthreads();

    if (tid < 32) {
        float ssum = 0.f;
#pragma unroll
        for (int j = 0; j < 32; ++j) {
            if (j != tid) ssum += expf(logits[tid][j]);
        }
        float m = logf(ssum * (1.0f / 31.0f));
        bool valid = (m == m);  // !isnan
        float contrib = valid ? m : 0.f;
        float cnt = valid ? 1.f : 0.f;
#pragma unroll
        for (int off = 16; off; off >>= 1) {
            contrib += __shfl_xor(contrib, off, 32);
            cnt += __shfl_xor(cnt, off, 32);
        }
        if (tid == 0) out[0] = contrib / cnt;
    }
}

extern "C" void kernel_launch(void* const* d_in, const int* in_sizes, int n_in,
                              void* d_out, int out_size, void* d_ws,
                              size_t ws_size, hipStream_t stream) {
    const float* feats = (const float*)d_in[0];       // [BATCH, 512] f32
    const float* protos0 = (const float*)d_in[1];     // [32, 512] f32
    const int* labels = (const int*)d_in[2];          // [BATCH] i32
    float* out = (float*)d_out;                       // scalar f32
    const int batch = in_sizes[2];

    // Workspace: [0, batch*4)        : ordered per-state index lists (compact)
    //            [aligned, +64KB)    : final prototypes [32, 512] f32
    size_t list_bytes = ((size_t)batch * sizeof(int) + 255) & ~(size_t)255;
    int* list = (int*)d_ws;
    float* protos = (float*)((char*)d_ws + list_bytes);

    disloss_ema_kernel<<<N_STATES, 32, 0, stream>>>(feats, protos0, labels,
                                                    batch, list, protos);
    disloss_wmma_loss_kernel<<<1, 128, 0, stream>>>(protos, out);
}